// PtModule_76166950027882
// MI455X (gfx1250) — compile-verified
//
#include <hip/hip_runtime.h>

// ---------------------------------------------------------------------------
// MI455X (gfx1250) fused streaming kernel:
//   out_add = x + 2 + row ;  out_mul = x * 3 ;  mean(out_add)
// Bandwidth-bound: 768 MB traffic -> ~33us floor @ 23.3 TB/s.
// 128-bit NT loads/stores; cross-lane mean reduction via V_WMMA_F32_16X16X4_F32.
// ---------------------------------------------------------------------------

typedef float f32x4 __attribute__((ext_vector_type(4)));
typedef float v2f   __attribute__((ext_vector_type(2)));
typedef float v8f   __attribute__((ext_vector_type(8)));

#define N_ELEMS    67108864u   // 8192 * 8192
#define N_VEC4     16777216u   // N_ELEMS / 4
#define ROW_SHIFT  11          // vec4-index >> 11 == row (2048 vec4 per 8192-f32 row)
#define BLOCKS     8192
#define THREADS    256         // 8 wave32 per block
#define ITERS      (N_VEC4 / (BLOCKS * THREADS))   // == 8, exact: no tail divergence

// Full-wave (32 lane) sum using one f32 WMMA.
// A(16x4) per ISA layout: lane L<16 holds {M=L,K=0 ; K=1}, lane L>=16 holds {K=2 ; K=3}.
// With A = {partial, 0} and B = ones(4x16):
//   D[m][n] = partial(lane m) + partial(lane m+16)   for every n
// so sum(d[0..7]) is uniform within each 16-lane half; one xor-16 shuffle
// produces the full 32-lane sum on every lane.
__device__ __forceinline__ float wave_sum_wmma_f32(float partial) {
  v2f a; a.x = partial; a.y = 0.0f;
  v2f b; b.x = 1.0f;    b.y = 1.0f;
  v8f c = {};
  v8f d = __builtin_amdgcn_wmma_f32_16x16x4_f32(
      /*neg_a=*/false, a, /*neg_b=*/false, b,
      /*c_mod=*/(short)0, c, /*reuse_a=*/false, /*reuse_b=*/false);
  float s = (d[0] + d[1]) + (d[2] + d[3]) + (d[4] + d[5]) + (d[6] + d[7]);
  return s + __shfl_xor(s, 16, 32);
}

__global__ void __launch_bounds__(THREADS)
fused_stream_kernel(const float* __restrict__ x,
                    float* __restrict__ out_add,
                    float* __restrict__ out_mul,
                    float* __restrict__ partials) {
  const unsigned tid    = blockIdx.x * THREADS + threadIdx.x;
  const unsigned stride = BLOCKS * THREADS;

  const f32x4* __restrict__ xv = reinterpret_cast<const f32x4*>(x);
  f32x4* __restrict__ av       = reinterpret_cast<f32x4*>(out_add);
  f32x4* __restrict__ mv       = reinterpret_cast<f32x4*>(out_mul);

  float acc = 0.0f;
#pragma unroll
  for (unsigned k = 0; k < ITERS; ++k) {
    const unsigned i = tid + k * stride;               // coalesced b128 accesses
    f32x4 v = __builtin_nontemporal_load(&xv[i]);      // x streamed once (256MB > L2)
    const float row = (float)(i >> ROW_SHIFT);         // rows <= 8191: exact in f32
    f32x4 add = v + (2.0f + row);
    f32x4 mul = v * 3.0f;
    __builtin_nontemporal_store(add, &av[i]);          // streaming stores, no L2 pollution
    __builtin_nontemporal_store(mul, &mv[i]);
    acc += (add.x + add.y) + (add.z + add.w);
  }

  // EXEC is all-ones here (uniform trip count, no divergence) -> WMMA legal.
  const float wsum = wave_sum_wmma_f32(acc);

  __shared__ float lds[THREADS / 32];
  const int lane = threadIdx.x & 31;
  const int wid  = threadIdx.x >> 5;
  if (lane == 0) lds[wid] = wsum;
  __syncthreads();
  if (threadIdx.x == 0) {
    float bs = 0.0f;
#pragma unroll
    for (int w = 0; w < THREADS / 32; ++w) bs += lds[w];
    partials[blockIdx.x] = bs;                         // deterministic, no atomics
  }
}

// One block folds the 8192 block-partials into the mean (double accumulation:
// the grand sum is ~2.7e11, f32-only accumulation would bias the mean).
__global__ void __launch_bounds__(256)
final_mean_kernel(const float* __restrict__ partials, float* __restrict__ mean_out) {
  double acc = 0.0;
  for (int i = threadIdx.x; i < BLOCKS; i += 256) acc += (double)partials[i];
#pragma unroll
  for (int off = 16; off >= 1; off >>= 1) acc += __shfl_xor(acc, off, 32);

  __shared__ double lds[8];
  const int lane = threadIdx.x & 31;
  const int wid  = threadIdx.x >> 5;
  if (lane == 0) lds[wid] = acc;
  __syncthreads();
  if (threadIdx.x == 0) {
    double t = 0.0;
#pragma unroll
    for (int w = 0; w < 8; ++w) t += lds[w];
    *mean_out = (float)(t / (double)N_ELEMS);
  }
}

extern "C" void kernel_launch(void* const* d_in, const int* in_sizes, int n_in,
                              void* d_out, int out_size, void* d_ws, size_t ws_size,
                              hipStream_t stream) {
  (void)in_sizes; (void)n_in; (void)out_size; (void)ws_size;

  const float* x = (const float*)d_in[0];
  float* out       = (float*)d_out;
  float* out_add   = out;                         // [0, N)
  float* out_mul   = out + (size_t)N_ELEMS;       // [N, 2N)
  float* mean_out  = out + (size_t)2 * N_ELEMS;   // [2N]
  float* partials  = (float*)d_ws;                // BLOCKS floats (32 KB scratch)

  fused_stream_kernel<<<BLOCKS, THREADS, 0, stream>>>(x, out_add, out_mul, partials);
  final_mean_kernel<<<1, 256, 0, stream>>>(partials, mean_out);
}